// MultiYoloLoss_42674795053769
// MI455X (gfx1250) — compile-verified
//
#include <hip/hip_runtime.h>
#include <hip/hip_bf16.h>

#define NUM_CLASSES 80
#define NPRED 22743          // 3*(19^2+38^2+76^2)
#define NGT   40
#define NBATCH 16
#define TPB 256

typedef float v2f __attribute__((ext_vector_type(2)));
typedef float v8f __attribute__((ext_vector_type(8)));

__constant__ float c_anchors[8][2] = {
    {10.f, 13.f}, {16.f, 30.f}, {33.f, 23.f}, {61.f, 62.f},
    {59.f, 119.f}, {116.f, 90.f}, {156.f, 198.f}, {373.f, 326.f}};
__constant__ int   c_scale_of_anchor[8]  = {2, 2, 2, 1, 1, 1, 0, 0};
__constant__ float c_stride_of_anchor[8] = {8.f, 8.f, 8.f, 16.f, 16.f, 16.f, 32.f, 32.f};
__constant__ int   c_grid_w[3] = {19, 38, 76};
__constant__ int   c_offs[3]   = {0, 1083, 5415};

__device__ __forceinline__ float sigmoidf(float x) { return 1.f / (1.f + expf(-x)); }
__device__ __forceinline__ float clip01(float p) {
    return fminf(fmaxf(p, 1e-7f), 1.f - 1e-7f);
}

// ------------------------------------------------------------------
// Kernel 1: per-GT preprocessing (640 GTs total).
// Record layout (12 floats): x1,y1,x2,y2, tx,ty,tw,th, scale_w,
//                            label(int bits), idx(int bits), area
// ------------------------------------------------------------------
__global__ void yolo_prep(const float* __restrict__ targets,
                          float* __restrict__ gtrec) {
    int i = blockIdx.x * blockDim.x + threadIdx.x;
    if (i >= NBATCH * NGT) return;
    const float* t = targets + (size_t)i * 5;
    float x1 = t[0] * 608.f, y1 = t[1] * 608.f;
    float x2 = t[2] * 608.f, y2 = t[3] * 608.f;
    float label = t[4];
    float cx = 0.5f * (x1 + x2), cy = 0.5f * (y1 + y2);
    float gw = x2 - x1, gh = y2 - y1;

    int best = 0;
    float bestr = -1.f;
#pragma unroll
    for (int k = 0; k < 8; ++k) {
        float aw = c_anchors[k][0], ah = c_anchors[k][1];
        float inter = fminf(gw, aw) * fminf(gh, ah);
        float uni = gw * gh + aw * ah - inter;
        float r = inter / (uni + 1e-9f);
        if (r > bestr) { bestr = r; best = k; }
    }
    int   scale  = c_scale_of_anchor[best];
    float stride = c_stride_of_anchor[best];
    int   a_in   = best - (best / 3) * 3;
    int   wg     = c_grid_w[scale];
    int   off    = c_offs[scale];

    int gi = (int)floorf(cx / stride);
    int gj = (int)floorf(cy / stride);
    gi = min(max(gi, 0), wg - 1);
    gj = min(max(gj, 0), wg - 1);
    int idx = off + (gj * wg + gi) * 3 + a_in;

    float tx = cx / stride - (float)gi;
    float ty = cy / stride - (float)gj;
    float tw = logf(gw / c_anchors[best][0] + 1e-9f);
    float th = logf(gh / c_anchors[best][1] + 1e-9f);
    float scale_w = 2.f - gw * gh * (1.f / (608.f * 608.f));

    float* r = gtrec + (size_t)i * 12;
    r[0] = x1; r[1] = y1; r[2] = x2; r[3] = y2;
    r[4] = tx; r[5] = ty; r[6] = tw; r[7] = th;
    r[8] = scale_w;
    r[9]  = __int_as_float((int)label);
    r[10] = __int_as_float(idx);
    r[11] = gw * gh;
}

// ------------------------------------------------------------------
// Kernel 2: main per-prediction loss. grid = (ceil(NPRED/TPB), B).
// Work index w enumerates (scale, anchor, cell) with cell fastest
// so channel loads are coalesced. Only channels 0..4 are read unless
// the prediction is foreground (then 80 class channels too).
// ------------------------------------------------------------------
__global__ void __launch_bounds__(TPB)
yolo_main(const float* __restrict__ ldat, const float* __restrict__ mdat,
          const float* __restrict__ hdat, const float* __restrict__ gtrec,
          float* __restrict__ partials, int nblk_x) {
    const int b   = blockIdx.y;
    const int tid = threadIdx.x;
    const int w   = blockIdx.x * TPB + tid;

    __shared__ float s_gt[NGT * 12];
    __shared__ float s_red[TPB];

    for (int i = tid; i < NGT * 12; i += TPB)
        s_gt[i] = gtrec[(size_t)b * (NGT * 12) + i];
    __syncthreads();

    float loss = 0.f;
    if (w < NPRED) {
        const float* dat;
        int HW, W, a, p, off, abase;
        float fstride;
        if (w < 1083) {                       // 19x19, stride 32, anchors 6..8
            int rem = w;      a = rem / 361;  p = rem - a * 361;
            HW = 361;  W = 19; dat = ldat; fstride = 32.f; abase = 6; off = 0;
        } else if (w < 5415) {                // 38x38, stride 16, anchors 3..5
            int rem = w - 1083; a = rem / 1444; p = rem - a * 1444;
            HW = 1444; W = 38; dat = mdat; fstride = 16.f; abase = 3; off = 1083;
        } else {                              // 76x76, stride 8, anchors 0..2
            int rem = w - 5415; a = rem / 5776; p = rem - a * 5776;
            HW = 5776; W = 76; dat = hdat; fstride = 8.f;  abase = 0; off = 5415;
        }
        const float* base = dat + ((size_t)b * 255 + (size_t)a * 85) * HW + p;
        float t0 = base[0];
        float t1 = base[(size_t)HW];
        float t2 = base[2 * (size_t)HW];
        float t3 = base[3 * (size_t)HW];
        float t4 = base[4 * (size_t)HW];

        float sx = sigmoidf(t0), sy = sigmoidf(t1);
        float conf = sigmoidf(t4);

        int gi = p % W, gj = p / W;
        float aw = c_anchors[abase + a][0], ah = c_anchors[abase + a][1];

        // decode
        float bx = (sx + (float)gi) * fstride;
        float by = (sy + (float)gj) * fstride;
        float bw = expf(t2) * aw;
        float bh = expf(t3) * ah;
        float ax1 = bx - 0.5f * bw, ay1 = by - 0.5f * bh;
        float ax2 = bx + 0.5f * bw, ay2 = by + 0.5f * bh;
        float area_a = (ax2 - ax1) * (ay2 - ay1);

        int n_ref = off + p * 3 + a;   // index in the reference pred ordering

        float max_iou = -1.f;
        int tmatch = -1;
#pragma unroll 4
        for (int t = 0; t < NGT; ++t) {
            const float* r = &s_gt[t * 12];
            float tlx = fmaxf(ax1, r[0]);
            float tly = fmaxf(ay1, r[1]);
            float brx = fminf(ax2, r[2]);
            float bry = fminf(ay2, r[3]);
            float iw = fmaxf(brx - tlx, 0.f);
            float ih = fmaxf(bry - tly, 0.f);
            float inter = iw * ih;
            float iou = inter / (area_a + r[11] - inter + 1e-9f);
            max_iou = fmaxf(max_iou, iou);
            if (__float_as_int(r[10]) == n_ref) tmatch = t;  // last match wins
        }

        if (tmatch >= 0) {
            const float* r = &s_gt[tmatch * 12];
            // localization
            float dx = sx - r[4], dy = sy - r[5];
            float dw = t2 - r[6], dh = t3 - r[7];
            loss += r[8] * 0.5f * (dx * dx + dy * dy + dw * dw + dh * dh);
            // objectness (target 1)
            loss += -logf(clip01(conf));
            // classes — only now touch the 80 class channels
            const float* cbase = base + 5 * (size_t)HW;
            __builtin_prefetch(cbase, 0, 0);   // global_prefetch_b8
            int lab = __float_as_int(r[9]);
            for (int c = 0; c < NUM_CLASSES; ++c) {
                float pc = clip01(sigmoidf(cbase[(size_t)c * HW]));
                loss += (c == lab) ? -logf(pc) : -log1pf(-pc);
            }
        } else if (max_iou < 0.5f) {
            // background objectness (target 0)
            loss += -log1pf(-clip01(conf));
        }
    }

    // block reduction (deterministic tree)
    s_red[tid] = loss;
    __syncthreads();
#pragma unroll
    for (int st = TPB / 2; st > 0; st >>= 1) {
        if (tid < st) s_red[tid] += s_red[tid + st];
        __syncthreads();
    }
    if (tid == 0) partials[(size_t)b * nblk_x + blockIdx.x] = s_red[0];
}

// ------------------------------------------------------------------
// Kernel 3: single-wave final reduction using V_WMMA_F32_16X16X4_F32.
// A = lane partial sums (K0 slot), B = ones -> D[m][n] = s_m + s_{m+16}.
// Summing a lane's 8 accumulator VGPRs gives half the total; one
// shfl_xor(16) completes the exact f32 sum.
// ------------------------------------------------------------------
__global__ void yolo_reduce(const float* __restrict__ partials, int P,
                            float* __restrict__ out) {
    int lane = threadIdx.x;  // 32 threads, one wave, EXEC all ones
    float s = 0.f;
    for (int i = lane; i < P; i += 32) s += partials[i];

    v2f A;  A.x = s;   A.y = 0.f;
    v2f Bn; Bn.x = 1.f; Bn.y = 1.f;
    v8f C = {};
    C = __builtin_amdgcn_wmma_f32_16x16x4_f32(
        /*neg_a=*/false, A, /*neg_b=*/false, Bn,
        /*c_mod=*/(short)0, C, /*reuse_a=*/false, /*reuse_b=*/false);

    float t = C[0] + C[1] + C[2] + C[3] + C[4] + C[5] + C[6] + C[7];
    t += __shfl_xor(t, 16, 32);
    if (lane == 0) out[0] = t * (1.0f / (float)NBATCH);
}

// ------------------------------------------------------------------
extern "C" void kernel_launch(void* const* d_in, const int* in_sizes, int n_in,
                              void* d_out, int out_size, void* d_ws, size_t ws_size,
                              hipStream_t stream) {
    const float* l  = (const float*)d_in[0];
    const float* m  = (const float*)d_in[1];
    const float* h  = (const float*)d_in[2];
    const float* tg = (const float*)d_in[3];
    float* out = (float*)d_out;

    float* gtrec    = (float*)d_ws;                      // 640*12 floats
    float* partials = (float*)((char*)d_ws + 32768);     // nblk*16 floats

    int nblk = (NPRED + TPB - 1) / TPB;                  // 89

    yolo_prep<<<(NBATCH * NGT + 255) / 256, 256, 0, stream>>>(tg, gtrec);

    dim3 grid(nblk, NBATCH);
    yolo_main<<<grid, TPB, 0, stream>>>(l, m, h, gtrec, partials, nblk);

    yolo_reduce<<<1, 32, 0, stream>>>(partials, nblk * NBATCH, out);
}